// LightGCN_84902913507819
// MI455X (gfx1250) — compile-verified
//
#include <hip/hip_runtime.h>
#include <stdint.h>

#define USER_NUM 200000
#define ITEM_NUM 100000
#define N_NODES  (USER_NUM + ITEM_NUM)
#define DIM      64
#define WPB      8      // waves per block (256 threads, wave32)

// 16-byte LDS record per edge: {src, dst, w_bits, pad}
typedef unsigned int u32x4 __attribute__((ext_vector_type(4)));
typedef volatile __attribute__((address_space(3))) u32x4 lds_u32x4_t;

// ---------------------------------------------------------------------------
// out[i] = cur[i] = concat(user_emb, item_emb)[i]   (float4 vectorized)
// ---------------------------------------------------------------------------
__global__ void lg_init(const float4* __restrict__ ue, const float4* __restrict__ ie,
                        float4* __restrict__ cur, float4* __restrict__ acc,
                        int nUser4, int n4) {
    int i = blockIdx.x * blockDim.x + threadIdx.x;
    if (i < n4) {
        float4 v = (i < nUser4) ? ue[i] : ie[i - nUser4];
        cur[i] = v;
        acc[i] = v;
    }
}

__global__ void lg_zero(float4* __restrict__ p, int n4) {
    int i = blockIdx.x * blockDim.x + threadIdx.x;
    if (i < n4) p[i] = make_float4(0.f, 0.f, 0.f, 0.f);
}

// acc = (acc + cur) * s   — s=1 for inner layers, s=0.25 folds the final mean
__global__ void lg_accscale(float4* __restrict__ acc, const float4* __restrict__ cur,
                            float s, int n4) {
    int i = blockIdx.x * blockDim.x + threadIdx.x;
    if (i < n4) {
        float4 a = acc[i], c = cur[i];
        a.x = (a.x + c.x) * s;
        a.y = (a.y + c.y) * s;
        a.z = (a.z + c.z) * s;
        a.w = (a.w + c.w) * s;
        acc[i] = a;
    }
}

// ---------------------------------------------------------------------------
// SpMM: y[dst] += w * x[src] over COO edges.
// One wave per 32-edge batch. Edge triples double-buffered into LDS with the
// CDNA5 async-to-LDS engine (ASYNCcnt), staged INTERLEAVED (16B/edge) so the
// inner loop reads each edge's metadata with one ds_load_b128 broadcast.
// Gathers are 256B/wave coalesced loads (L2-resident: node table fits in
// MI455X's 192MB L2); scatter uses HW global_atomic_add_f32.
// ---------------------------------------------------------------------------
__global__ __launch_bounds__(256) void lg_spmm(const float* __restrict__ x,
                                               float* __restrict__ y,
                                               const int* __restrict__ esrc,
                                               const int* __restrict__ edst,
                                               const float* __restrict__ ew,
                                               int E) {
    // [buf][wave][edge] : one 16B record per edge
    __shared__ u32x4 stg[2][WPB][32];

    const int lane = threadIdx.x & 31;
    const int wid  = threadIdx.x >> 5;
    const long long gwave  = ((long long)blockIdx.x * blockDim.x + threadIdx.x) >> 5;
    const long long nwaves = ((long long)gridDim.x * blockDim.x) >> 5;
    const long long nBatch = ((long long)E + 31) >> 5;

    // 32-bit LDS byte offsets (low 32 bits of a generic pointer into LDS).
    uint32_t base[2];   // &stg[b][wid][0]
    uint32_t lA[2], lB[2], lC[2];
    #pragma unroll
    for (int b = 0; b < 2; ++b) {
        base[b] = (uint32_t)(uintptr_t)&stg[b][wid][0];
        lA[b]   = base[b] + 16u * (unsigned)lane + 0u;   // src
        lB[b]   = base[b] + 16u * (unsigned)lane + 4u;   // dst
        lC[b]   = base[b] + 16u * (unsigned)lane + 8u;   // w bits
    }

    auto stage = [&](long long b, int buf) {
        long long e = (b << 5) + lane;
        if (e < (long long)E) {
            unsigned long long a0 = (unsigned long long)(uintptr_t)(esrc + e);
            unsigned long long a1 = (unsigned long long)(uintptr_t)(edst + e);
            unsigned long long a2 = (unsigned long long)(uintptr_t)(ew   + e);
            asm volatile("global_load_async_to_lds_b32 %0, %1, off"
                         :: "v"(lA[buf]), "v"(a0) : "memory");
            asm volatile("global_load_async_to_lds_b32 %0, %1, off"
                         :: "v"(lB[buf]), "v"(a1) : "memory");
            asm volatile("global_load_async_to_lds_b32 %0, %1, off"
                         :: "v"(lC[buf]), "v"(a2) : "memory");
        }
    };

    int buf = 0;
    long long b = gwave;
    if (b < nBatch) stage(b, 0);

    for (; b < nBatch; b += nwaves) {
        const long long nb = b + nwaves;
        const bool hasNext = nb < nBatch;
        if (hasNext) stage(nb, buf ^ 1);

        // Prefetch edge stream two strides ahead into L2 (global_prefetch_b8)
        {
            long long pe = ((b + 2 * nwaves) << 5) + lane;
            if (pe < (long long)E) {
                __builtin_prefetch(esrc + pe, 0, 2);
                __builtin_prefetch(edst + pe, 0, 2);
                __builtin_prefetch(ew   + pe, 0, 2);
            }
        }

        // Async loads complete in order: <=3 outstanding releases the current
        // buffer while the next batch's 3 loads remain in flight.
        if (hasNext) asm volatile("s_wait_asynccnt 3" ::: "memory");
        else         asm volatile("s_wait_asynccnt 0" ::: "memory");

        long long rem = (long long)E - (b << 5);
        int cnt = rem < 32 ? (int)rem : 32;

        // AS(3) typed pointer -> one ds_load_b128 broadcast per edge record.
        lds_u32x4_t* vp = (lds_u32x4_t*)base[buf];

        for (int j = 0; j < cnt; ++j) {
            u32x4 t = vp[j];
            unsigned int s = t.x;            // src node
            unsigned int d = t.y;            // dst node
            float        w = __uint_as_float(t.z);

            // Coalesced 256B wave gather of x[src][0..63], 2 floats/lane
            const float2 v = *((const float2*)(x + (size_t)s * DIM) + lane);

            float* yp = y + (size_t)d * DIM + (lane << 1);
            unsafeAtomicAdd(yp,     w * v.x);   // global_atomic_add_f32
            unsafeAtomicAdd(yp + 1, w * v.y);
        }
        buf ^= 1;
    }
}

// ---------------------------------------------------------------------------
extern "C" void kernel_launch(void* const* d_in, const int* in_sizes, int n_in,
                              void* d_out, int out_size, void* d_ws, size_t ws_size,
                              hipStream_t stream) {
    const float* ue   = (const float*)d_in[0];
    const float* ie   = (const float*)d_in[1];
    const float* ew   = (const float*)d_in[2];
    const int*   esrc = (const int*)d_in[3];
    const int*   edst = (const int*)d_in[4];
    const int E = in_sizes[2];

    float* out = (float*)d_out;                       // acc lives in d_out
    float* A   = (float*)d_ws;                        // layer buffers (ping-pong)
    float* B   = A + (size_t)N_NODES * DIM;           // 2 x 76.8 MB in ws

    const int n4     = N_NODES * DIM / 4;
    const int nUser4 = USER_NUM * DIM / 4;
    dim3 blk(256);
    dim3 grdE((n4 + 255) / 256);
    const int spmmBlocks = 2048;                      // 16384 waves; ~18 batches each

    lg_init<<<grdE, blk, 0, stream>>>((const float4*)ue, (const float4*)ie,
                                      (float4*)A, (float4*)out, nUser4, n4);

    // Layer 1: A -> B
    lg_zero<<<grdE, blk, 0, stream>>>((float4*)B, n4);
    lg_spmm<<<spmmBlocks, blk, 0, stream>>>(A, B, esrc, edst, ew, E);
    lg_accscale<<<grdE, blk, 0, stream>>>((float4*)out, (const float4*)B, 1.0f, n4);

    // Layer 2: B -> A
    lg_zero<<<grdE, blk, 0, stream>>>((float4*)A, n4);
    lg_spmm<<<spmmBlocks, blk, 0, stream>>>(B, A, esrc, edst, ew, E);
    lg_accscale<<<grdE, blk, 0, stream>>>((float4*)out, (const float4*)A, 1.0f, n4);

    // Layer 3: A -> B, fold the /4 mean into the final accumulate
    lg_zero<<<grdE, blk, 0, stream>>>((float4*)B, n4);
    lg_spmm<<<spmmBlocks, blk, 0, stream>>>(A, B, esrc, edst, ew, E);
    lg_accscale<<<grdE, blk, 0, stream>>>((float4*)out, (const float4*)B, 0.25f, n4);
}